// NeSSDetector_35862976922253
// MI455X (gfx1250) — compile-verified
//
#include <hip/hip_runtime.h>
#include <stdint.h>

// ---------------- problem constants (fixed by reference setup_inputs) -------
constexpr int kB = 8;
constexpr int kH = 1024;
constexpr int kW = 1536;
constexpr int kHW = kH * kW;
constexpr int kN = 2048;        // top-k
constexpr int kBorder = 16;

// NMS tiling
constexpr int TILE_H = 32;
constexpr int TILE_W = 128;
constexpr int LTH = TILE_H + 2;     // 34 rows in LDS (halo)
constexpr int LTW = TILE_W + 2;     // 130 cols in LDS (halo)

constexpr int NBIN = 65536;         // 16-bit radix histogram
constexpr int CAP  = 393216;        // per-batch extrema capacity (HW/4)
constexpr int CANDCAP = 4096;       // per-batch candidate buffer (>= kN + ties)
constexpr int STAGE = 2048;         // per-WG LDS staging (expected ~455 extrema)

// ---------------- workspace layout (bytes) ----------------------------------
constexpr size_t OFF_HIST_HI = 0;
constexpr size_t OFF_HIST_LO = (size_t)kB * NBIN * 4;              // 2 MiB
constexpr size_t OFF_CTRL    = 2 * (size_t)kB * NBIN * 4;          // 4 MiB
constexpr size_t OFF_CAND    = OFF_CTRL + 4096;                    // 8B aligned
constexpr size_t OFF_EXT     = OFF_CAND + (size_t)kB * CANDCAP * 8;
constexpr size_t ZERO_BYTES  = OFF_CTRL + 256;
// ctrl u32 indices: [0..7]=ext_count [8..15]=T_hi [16..23]=cntAbove_hi
//                   [24..31]=Kfull   [32..39]=cand_count [48..55]=fallback flag

// ---------------- TDM builtin plumbing --------------------------------------
typedef unsigned int u32x4 __attribute__((ext_vector_type(4)));
typedef int          i32x4 __attribute__((ext_vector_type(4)));
typedef int          i32x8 __attribute__((ext_vector_type(8)));

#if __has_builtin(__builtin_amdgcn_tensor_load_to_lds)
#define HAS_TDM 1
#else
#define HAS_TDM 0
#endif

__device__ __forceinline__ void wait_tensorcnt0() {
#if __has_builtin(__builtin_amdgcn_s_wait_tensorcnt)
  __builtin_amdgcn_s_wait_tensorcnt(0);
#else
  asm volatile("s_wait_tensorcnt 0" ::: "memory");
#endif
}

// ---------------- Kernel A: TDM tile load + NMS + ness + hist + compaction --
__global__ __launch_bounds__(256) void ness_extrema_kernel(
    const float* __restrict__ score, const float* __restrict__ neur,
    unsigned* __restrict__ hist_hi, unsigned* __restrict__ ctrl,
    unsigned long long* __restrict__ ext) {
  __shared__ float tile[LTH * LTW];
  __shared__ unsigned long long stage[STAGE];
  __shared__ unsigned lcnt;
  __shared__ unsigned gbase;

  const int batch = blockIdx.z;
  const int row0  = blockIdx.y * TILE_H;
  const int col0  = blockIdx.x * TILE_W;
  int sr = row0 - 1; if (sr < 0) sr = 0; if (sr > kH - LTH) sr = kH - LTH;
  int sc = col0 - 1; if (sc < 0) sc = 0; if (sc > kW - LTW) sc = kW - LTW;
  const float* gsrc = score + (size_t)batch * kHW + (size_t)sr * kW + sc;

  if (threadIdx.x == 0) lcnt = 0u;

#if HAS_TDM
  // TDM ignores EXEC, so a lane-divergent guard could issue duplicate DMAs
  // from waves whose EXEC=0 region is not branch-skipped. Use a wave-uniform
  // SGPR condition -> s_cmp + s_cbranch: only wave 0 ever reaches the op.
  const unsigned waveId =
      (unsigned)__builtin_amdgcn_readfirstlane((int)(threadIdx.x >> 5));
  if (waveId == 0u) {
    // gfx1250: low 32 bits of a flat shared-memory address == LDS byte offset
    unsigned lds_addr = (unsigned)(uintptr_t)tile;
    unsigned long long ga = (unsigned long long)(uintptr_t)gsrc;

    u32x4 g0;
    g0[0] = 1u;                                           // count=1, user mode
    g0[1] = lds_addr;                                     // lds_addr [63:32]
    g0[2] = (unsigned)(ga & 0xFFFFFFFFull);               // global_addr lo
    g0[3] = (unsigned)((ga >> 32) & 0x01FFFFFFull) | (2u << 30); // hi + type=2

    // group1: data_size=4B(code 2), tensor 1536x1024, tile 130x34, stride=1536
    unsigned long long q0 = (2ull << 16)
                          | ((unsigned long long)(kW & 0xFFFF) << 48);
    unsigned long long q1 = (unsigned long long)((unsigned)kW >> 16)
                          | ((unsigned long long)kH << 16)
                          | ((unsigned long long)LTW << 48);
    unsigned long long q2 = (unsigned long long)LTH
                          | ((unsigned long long)kW << 32);
    unsigned long long q3 = 0ull;
    i32x8 g1;
    g1[0] = (int)(unsigned)q0; g1[1] = (int)(unsigned)(q0 >> 32);
    g1[2] = (int)(unsigned)q1; g1[3] = (int)(unsigned)(q1 >> 32);
    g1[4] = (int)(unsigned)q2; g1[5] = (int)(unsigned)(q2 >> 32);
    g1[6] = (int)(unsigned)q3; g1[7] = (int)(unsigned)(q3 >> 32);
    i32x4 gz = {0, 0, 0, 0};
#if defined(__clang_major__) && (__clang_major__ >= 23)
    i32x8 gz8 = {0, 0, 0, 0, 0, 0, 0, 0};
    __builtin_amdgcn_tensor_load_to_lds(g0, g1, gz, gz, gz8, 0);
#else
    __builtin_amdgcn_tensor_load_to_lds(g0, g1, gz, gz, 0);
#endif
    wait_tensorcnt0();
  }
#else
  for (int p = threadIdx.x; p < LTH * LTW; p += 256)
    tile[p] = gsrc[(p / LTW) * kW + (p % LTW)];
#endif
  __syncthreads();

  const size_t bbase = (size_t)batch * kHW;
  for (int p = threadIdx.x; p < TILE_H * TILE_W; p += 256) {
    const int ly = p / TILE_W, lx = p % TILE_W;
    const int y = row0 + ly, x = col0 + lx;
    if (y < kBorder || y >= kH - kBorder || x < kBorder || x >= kW - kBorder)
      continue;
    const float* c = &tile[(y - sr) * LTW + (x - sc)];
    const float s = c[0];
    // score == 3x3 pooled max  <=>  s >= all 8 neighbors
    bool m = (s >= c[-LTW - 1]) & (s >= c[-LTW]) & (s >= c[-LTW + 1])
           & (s >= c[-1])       & (s >= c[1])
           & (s >= c[LTW - 1])  & (s >= c[LTW]) & (s >= c[LTW + 1]);
    if (!m) continue;
    const int flat = y * kW + x;
    float nv = neur[bbase + flat];
    nv = fmaxf(nv, 0.0f);
    const float ness = expf(-nv);                  // in (0,1]
    const unsigned key = __float_as_uint(ness);    // monotone for positives
    // pack: key desc, then ~flat desc == flat asc (stable-topk tie-break)
    const unsigned long long packed =
        ((unsigned long long)key << 32) | (unsigned)(~flat);
    const unsigned lpos = atomicAdd(&lcnt, 1u);    // cheap LDS atomic
    if (lpos < (unsigned)STAGE) {
      stage[lpos] = packed;
    } else {                                       // rare overflow: direct path
      const unsigned gpos = atomicAdd(&ctrl[batch], 1u);
      if (gpos < (unsigned)CAP) {
        ext[(size_t)batch * CAP + gpos] = packed;
        atomicAdd(&hist_hi[(size_t)batch * NBIN + (key >> 16)], 1u);
      }
    }
  }
  __syncthreads();

  unsigned cnt = lcnt;
  if (cnt > (unsigned)STAGE) cnt = STAGE;
  if (threadIdx.x == 0) gbase = atomicAdd(&ctrl[batch], cnt); // 1 global atomic/WG
  __syncthreads();
  const unsigned base = gbase;
  for (unsigned i = threadIdx.x; i < cnt; i += 256) {
    const unsigned long long v = stage[i];
    const unsigned gpos = base + i;
    if (gpos < (unsigned)CAP) {
      ext[(size_t)batch * CAP + gpos] = v;
      atomicAdd(&hist_hi[(size_t)batch * NBIN + ((unsigned)(v >> 32) >> 16)], 1u);
    }
  }
}

// ---------------- Kernel B/D: radix threshold selection ----------------------
__global__ __launch_bounds__(1024) void select_kernel(
    const unsigned* __restrict__ hist, unsigned* __restrict__ ctrl, int mode) {
  const int batch = blockIdx.x;
  const unsigned* h = hist + (size_t)batch * NBIN;
  unsigned target;
  if (mode == 0) {
    target = (unsigned)kN;
  } else {
    unsigned ca = ctrl[16 + batch];
    target = (ca < (unsigned)kN) ? (unsigned)kN - ca : 1u;
  }

  __shared__ unsigned chunkSum[1024];
  __shared__ unsigned sufAfter[1024];
  const int t = threadIdx.x;
  unsigned s = 0;
  const int base = t * (NBIN / 1024);
  for (int b = 0; b < NBIN / 1024; ++b) s += h[base + b];
  chunkSum[t] = s;
  __syncthreads();
  if (t == 0) {
    unsigned run = 0;
    for (int c2 = 1023; c2 >= 0; --c2) { sufAfter[c2] = run; run += chunkSum[c2]; }
    if (run < target) {                    // fewer items than needed: take all
      if (mode == 0) { ctrl[8 + batch] = 0; ctrl[16 + batch] = run; }
      else           { ctrl[24 + batch] = (ctrl[8 + batch] << 16); }
      ctrl[48 + batch] = 1u;
    } else {
      ctrl[48 + batch] = 0u;
    }
  }
  __syncthreads();
  if (ctrl[48 + batch]) return;

  unsigned cum = sufAfter[t];
  for (int b = NBIN / 1024 - 1; b >= 0; --b) {
    const unsigned c2 = cum + h[base + b];
    if (cum < target && c2 >= target) {    // unique crossing point
      if (mode == 0) {
        ctrl[8 + batch]  = (unsigned)(base + b);  // high-16 threshold bin
        ctrl[16 + batch] = cum;                   // #items strictly above
      } else {
        ctrl[24 + batch] = (ctrl[8 + batch] << 16) | (unsigned)(base + b);
      }
    }
    cum = c2;
  }
}

// ---------------- Kernel C: low-16 histogram over threshold bin --------------
__global__ __launch_bounds__(256) void hist_lo_kernel(
    const unsigned long long* __restrict__ ext, const unsigned* __restrict__ ctrl,
    unsigned* __restrict__ hist_lo) {
  const int batch = blockIdx.y;
  unsigned cnt = ctrl[batch];
  if (cnt > (unsigned)CAP) cnt = CAP;
  const unsigned T = ctrl[8 + batch];
  const unsigned stride = gridDim.x * blockDim.x;
  for (unsigned i = blockIdx.x * blockDim.x + threadIdx.x; i < cnt; i += stride) {
    const unsigned key = (unsigned)(ext[(size_t)batch * CAP + i] >> 32);
    if ((key >> 16) == T)
      atomicAdd(&hist_lo[(size_t)batch * NBIN + (key & 0xFFFFu)], 1u);
  }
}

// ---------------- Kernel E: compact candidates >= full threshold -------------
__global__ __launch_bounds__(256) void compact_kernel(
    const unsigned long long* __restrict__ ext, unsigned* __restrict__ ctrl,
    unsigned long long* __restrict__ cand) {
  const int batch = blockIdx.y;
  unsigned cnt = ctrl[batch];
  if (cnt > (unsigned)CAP) cnt = CAP;
  const unsigned Kfull = ctrl[24 + batch];
  const unsigned stride = gridDim.x * blockDim.x;
  for (unsigned i = blockIdx.x * blockDim.x + threadIdx.x; i < cnt; i += stride) {
    const unsigned long long v = ext[(size_t)batch * CAP + i];
    if ((unsigned)(v >> 32) >= Kfull) {
      const unsigned pos = atomicAdd(&ctrl[32 + batch], 1u);
      if (pos < (unsigned)CANDCAP) cand[(size_t)batch * CANDCAP + pos] = v;
    }
  }
}

// ---------------- Kernel F: per-batch bitonic sort + Taylor refinement -------
constexpr int SORTN = 4096;
__global__ __launch_bounds__(1024) void sort_finalize_kernel(
    const unsigned long long* __restrict__ cand, const unsigned* __restrict__ ctrl,
    const float* __restrict__ score, const float* __restrict__ neur,
    float* __restrict__ out) {
  __shared__ unsigned long long keys[SORTN];
  const int batch = blockIdx.x;
  unsigned m = ctrl[32 + batch];
  if (m > (unsigned)SORTN) m = SORTN;
  const int tid = threadIdx.x;
  for (int i = tid; i < SORTN; i += 1024)
    keys[i] = (i < (int)m) ? cand[(size_t)batch * CANDCAP + i] : 0ull;

  for (unsigned k = 2; k <= (unsigned)SORTN; k <<= 1)
    for (unsigned j = k >> 1; j > 0; j >>= 1) {
      __syncthreads();
      for (unsigned i = tid; i < (unsigned)SORTN; i += 1024) {
        const unsigned ixj = i ^ j;
        if (ixj > i) {
          const unsigned long long a = keys[i], b = keys[ixj];
          const bool up = ((i & k) == 0);      // descending overall
          if (up ? (a < b) : (a > b)) { keys[i] = b; keys[ixj] = a; }
        }
      }
    }
  __syncthreads();

  const float* sf = score + (size_t)batch * kHW;
  const float* nf = neur  + (size_t)batch * kHW;
  for (int i = tid; i < kN; i += 1024) {
    const unsigned long long v = keys[i];
    unsigned idx = ~(unsigned)(v & 0xFFFFFFFFull);
    if (idx >= (unsigned)kHW) idx = 0;
    const int y = (int)(idx / (unsigned)kW);
    const int x = (int)(idx % (unsigned)kW);
    const int yc = (y < 1) ? 1 : ((y > kH - 2) ? kH - 2 : y);
    const int xc = (x < 1) ? 1 : ((x > kW - 2) ? kW - 2 : x);
    const float* r = sf + (size_t)yc * kW + xc;
    __builtin_prefetch(r - kW - 1, 0, 0);
    const float s00 = r[0];
    const float sp0 = r[kW],      sm0 = r[-kW];
    const float s0p = r[1],       s0m = r[-1];
    const float spp = r[kW + 1],  spm = r[kW - 1];
    const float smp = r[-kW + 1], smm = r[-kW - 1];

    const float gy  = 0.5f * (sp0 - sm0);
    const float gx  = 0.5f * (s0p - s0m);
    const float hyy = sp0 - 2.0f * s00 + sm0;
    const float hxx = s0p - 2.0f * s00 + s0m;
    const float hxy = 0.25f * (spp - spm - smp + smm);
    const float det = hyy * hxx - hxy * hxy;
    const bool sing = fabsf(det) > 1e-12f;
    const float sd  = sing ? det : 1.0f;
    float iy = -(hxx * gy - hxy * gx) / sd;
    float ix = -(hyy * gx - hxy * gy) / sd;
    if (!sing) { iy = 0.0f; ix = 0.0f; }
    iy = fminf(fmaxf(iy, -0.5f), 0.5f);
    ix = fminf(fmaxf(ix, -0.5f), 0.5f);

    float* o = out + ((size_t)batch * kN + i) * 3;
    o[0] = (float)y + 0.5f + iy;
    o[1] = (float)x + 0.5f + ix;
    o[2] = fmaxf(nf[idx], 0.0f);
  }
}

// ---------------- host launcher ---------------------------------------------
extern "C" void kernel_launch(void* const* d_in, const int* in_sizes, int n_in,
                              void* d_out, int out_size, void* d_ws, size_t ws_size,
                              hipStream_t stream) {
  (void)in_sizes; (void)n_in; (void)out_size; (void)ws_size;
  const float* neur  = (const float*)d_in[0];
  const float* score = (const float*)d_in[1];
  float* out = (float*)d_out;

  char* ws = (char*)d_ws;
  unsigned* hist_hi = (unsigned*)(ws + OFF_HIST_HI);
  unsigned* hist_lo = (unsigned*)(ws + OFF_HIST_LO);
  unsigned* ctrl    = (unsigned*)(ws + OFF_CTRL);
  unsigned long long* cand = (unsigned long long*)(ws + OFF_CAND);
  unsigned long long* ext  = (unsigned long long*)(ws + OFF_EXT);

  hipMemsetAsync(d_ws, 0, ZERO_BYTES, stream);

  dim3 gridA(kW / TILE_W, kH / TILE_H, kB);                 // 12 x 32 x 8
  ness_extrema_kernel<<<gridA, 256, 0, stream>>>(score, neur, hist_hi, ctrl, ext);

  select_kernel<<<kB, 1024, 0, stream>>>(hist_hi, ctrl, 0); // high-16 threshold

  dim3 gridS(256, kB);
  hist_lo_kernel<<<gridS, 256, 0, stream>>>(ext, ctrl, hist_lo);

  select_kernel<<<kB, 1024, 0, stream>>>(hist_lo, ctrl, 1); // full 32-bit thr

  compact_kernel<<<gridS, 256, 0, stream>>>(ext, ctrl, cand);

  sort_finalize_kernel<<<kB, 1024, 0, stream>>>(cand, ctrl, score, neur, out);
}